// SMC_Transf_Cell_18313740550515
// MI455X (gfx1250) — compile-verified
//
#include <hip/hip_runtime.h>
#include <hip/hip_bf16.h>
#include <math.h>

// Problem dims
#define BB   16
#define PP   10
#define SS   128
#define DD   512
#define HH   8
#define DFFC 2048
#define VOC  32000
#define MR   (BB * PP)      // 160 rows for all GEMMs
#define SIGMA_C 0.05f

typedef __attribute__((ext_vector_type(2))) float v2f;
typedef __attribute__((ext_vector_type(8))) float v8f;

// Only check the target builtin in the DEVICE pass (host pass lacks AMDGCN builtins).
#if defined(__HIP_DEVICE_COMPILE__) && defined(__gfx1250__)
#if !__has_builtin(__builtin_amdgcn_wmma_f32_16x16x4_f32)
#error "gfx1250 device pass: __builtin_amdgcn_wmma_f32_16x16x4_f32 not available"
#endif
#endif

__device__ __forceinline__ v8f wmma_f32_16x16x4(v2f a, v2f b, v8f c) {
#if defined(__HIP_DEVICE_COMPILE__)
  return __builtin_amdgcn_wmma_f32_16x16x4_f32(
      false, a, false, b, (short)0, c, false, false);
#else
  (void)a; (void)b;
  return c;  // host pass: never executed, just needs to parse
#endif
}

// ---------------- Output layout (floats, concatenated in return order) -----
static constexpr size_t OUT3_OFF  = 0;                         // (B,P,1,D)   81920
static constexpr size_t ZOUT_OFF  = 81920;                     // (B,P,1,D)   81920
static constexpr size_t AVG_OFF   = 163840;                    // (B,1,VOC)  512000
static constexpr size_t MAXP_OFF  = 675840;                    // (B,VOC)    512000
static constexpr size_t EPS_OFF   = 1187840;                   // (B,P,1,D)   81920
static constexpr size_t ATTN_OFF  = 1269760;                   // (B,P,1,S)   20480
static constexpr size_t KOUT_OFF  = 1290240;                   // (B,P,S,D) 10485760
static constexpr size_t VOUT_OFF  = 11776000;                  // (B,P,S,D) 10485760
static constexpr size_t WOUT_OFF  = 22261760;                  // (B,P)         160
static constexpr size_t INEW_OFF  = 22261920;                  // (B,P,S)     20480

// ---------------- Workspace layout (floats) --------------------------------
static constexpr size_t WS_Q    = 0;         // 160x512   (later reused for ffn2)
static constexpr size_t WS_K    = 81920;     // 160x512
static constexpr size_t WS_V    = 163840;    // 160x512
static constexpr size_t WS_CTX  = 245760;    // 160x512
static constexpr size_t WS_Z    = 327680;    // 160x512
static constexpr size_t WS_O1   = 409600;    // 160x512
static constexpr size_t WS_F1   = 491520;    // 160x2048
static constexpr size_t WS_PRED = 819200;    // 160x32000
static constexpr size_t WS_W    = 5939200;   // 160
static constexpr size_t WS_IT   = 5939360;   // 160 (ints)

// ---------------- RNG (deterministic, graph-replay safe) -------------------
__device__ __forceinline__ unsigned pcg_hash(unsigned v) {
  unsigned s = v * 747796405u + 2891336453u;
  unsigned w = ((s >> ((s >> 28u) + 4u)) ^ s) * 277803737u;
  return (w >> 22u) ^ w;
}
__device__ __forceinline__ float rng_u01(unsigned h) {
  return ((float)(h >> 8) + 0.5f) * (1.0f / 16777216.0f);
}
__device__ __forceinline__ float rng_gauss(unsigned seed, unsigned idx) {
  unsigned h1 = pcg_hash(idx * 2u + seed * 0x9E3779B9u + 0x85EBCA6Bu);
  unsigned h2 = pcg_hash(idx * 2u + 1u + seed * 0xC2B2AE35u);
  float u1 = rng_u01(h1), u2 = rng_u01(h2);
  return sqrtf(-2.0f * __logf(u1)) * __cosf(6.28318530718f * u2);
}

// ---------------- f32 WMMA GEMM: C[M,N] = A[M,K] @ B[K,N] (+bias,relu) -----
// block = 128 threads = 4 waves. Each wave owns a 32(M) x 16(N) C strip as
// two 16x16 WMMA accumulators, so every B fragment (2 global loads) feeds
// TWO v_wmma_f32_16x16x4_f32 ops: halves B/L2 traffic vs one-tile-per-wave.
// A tile (32 x 64) staged in LDS per K-chunk with coalesced loads.
// B pointer is hoisted per K-chunk and strength-reduced (+= 4*N per step)
// to avoid a 64-bit multiply per fragment in the unrolled inner loop.
#define KCHUNK 64
__global__ __launch_bounds__(128) void gemm_f32_wmma(
    const float* __restrict__ A, const float* __restrict__ Bm,
    const float* __restrict__ bias, float* __restrict__ C,
    int M, int N, int K, int relu)
{
  __shared__ float As[32][KCHUNK + 4];
  const int tid  = threadIdx.x;
  const int wave = tid >> 5;
  const int lane = tid & 31;
  const int m0   = blockIdx.y * 32;
  const int n0   = blockIdx.x * 64 + wave * 16;
  const int half = (lane < 16) ? 0 : 2;     // K phase for A/B fragments
  const int mn   = lane & 15;               // row (A) / col (B,C) within tile
  const bool nok = (n0 + 16 <= N);          // wave-uniform
  const size_t strideN  = (size_t)N;        // floats per B row
  const size_t strideN4 = (size_t)N * 4;    // k advances by 4 per step

  // per-lane B pointer for (k = half, n = n0+mn); advanced by strideN4 per step
  const float* bptr = Bm + (size_t)half * strideN + (size_t)(n0 + mn);

  v8f acc0 = {};
  v8f acc1 = {};
  for (int kc = 0; kc < K; kc += KCHUNK) {
    for (int i = tid; i < 32 * KCHUNK; i += 128) {
      int r = i >> 6, c = i & (KCHUNK - 1);
      As[r][c] = (m0 + r < M) ? A[(size_t)(m0 + r) * K + kc + c] : 0.0f;
    }
    __syncthreads();
    if (nok) {
      const float* bp = bptr;
      #pragma unroll
      for (int kk = 0; kk < KCHUNK; kk += 4) {
        v2f a0, a1, b;
        a0.x = As[mn][kk + half];
        a0.y = As[mn][kk + half + 1];
        a1.x = As[mn + 16][kk + half];
        a1.y = As[mn + 16][kk + half + 1];
        b.x = bp[0];
        b.y = bp[strideN];
        bp += strideN4;
        acc0 = wmma_f32_16x16x4(a0, b, acc0);
        acc1 = wmma_f32_16x16x4(a1, b, acc1);
      }
    }
    __syncthreads();
    bptr += (size_t)KCHUNK * strideN;   // advance K-chunk base
  }
  if (!nok) return;
  const int n = n0 + mn;
  const float bv = bias ? bias[n] : 0.0f;
  #pragma unroll
  for (int j = 0; j < 8; ++j) {
    int m = m0 + j + ((lane < 16) ? 0 : 8);
    if (m < M) {
      float v = acc0[j] + bv;
      if (relu) v = fmaxf(v, 0.0f);
      C[(size_t)m * N + n] = v;
    }
    int m2 = m + 16;
    if (m2 < M) {
      float v = acc1[j] + bv;
      if (relu) v = fmaxf(v, 0.0f);
      C[(size_t)m2 * N + n] = v;
    }
  }
}

// ---------------- elementwise Gaussian noise (optionally export epsilon) ---
__global__ __launch_bounds__(256) void noise_kernel(
    float* __restrict__ buf, float* __restrict__ eps_out, int n, unsigned seed)
{
  int i = blockIdx.x * 256 + threadIdx.x;
  if (i >= n) return;
  float g = SIGMA_C * rng_gauss(seed, (unsigned)i);
  if (eps_out) eps_out[i] = g;
  buf[i] += g;
}

// ---------------- K/V gather along P + slice-t insertion (float4 stream) ---
__global__ __launch_bounds__(128) void gather_kv_kernel(
    const float* __restrict__ Kin, const float* __restrict__ Vin,
    const int* __restrict__ I, const int* __restrict__ tptr,
    const float* __restrict__ knew, const float* __restrict__ vnew,
    float* __restrict__ Kout, float* __restrict__ Vout)
{
  const int t   = *tptr;
  const int bps = blockIdx.x;        // (b*P+p)*S + s
  const int s   = bps % SS;
  const int bp  = bps / SS;
  const int b   = bp / PP;
  const int d4  = threadIdx.x;       // float4 index, 0..127
  float4 kq, vq;
  if (s == t) {
    kq = ((const float4*)(knew + (size_t)bp * DD))[d4];
    vq = ((const float4*)(vnew + (size_t)bp * DD))[d4];
  } else {
    const int ip = I[bps];
    const size_t src = (((size_t)b * PP + ip) * SS + s) * DD;
    kq = ((const float4*)(Kin + src))[d4];
    vq = ((const float4*)(Vin + src))[d4];
  }
  ((float4*)(Kout + (size_t)bps * DD))[d4] = kq;
  ((float4*)(Vout + (size_t)bps * DD))[d4] = vq;
}

// ---------------- per-(b,p) 8-head attention (q length 1) ------------------
__global__ __launch_bounds__(128) void attention_kernel(
    const float* __restrict__ q, const float* __restrict__ Kt,
    const float* __restrict__ Vt, float* __restrict__ ctx,
    float* __restrict__ attnw)
{
  const int bp  = blockIdx.x;
  const int tid = threadIdx.x;       // s = tid
  __shared__ float sq[DD];
  __shared__ float sc[HH][SS];
  __shared__ float shmax[HH], shsum[HH];

  for (int i = tid; i < DD; i += 128) sq[i] = q[(size_t)bp * DD + i];
  __syncthreads();

  const float* Krow = Kt + ((size_t)bp * SS + tid) * DD;
  for (int h = 0; h < HH; ++h) {
    float dot = 0.f;
    #pragma unroll 8
    for (int d = 0; d < 64; ++d) dot += sq[h * 64 + d] * Krow[h * 64 + d];
    sc[h][tid] = dot * 0.125f;       // 1/sqrt(64)
  }
  __syncthreads();
  if (tid < HH) {
    float mx = -3.4e38f;
    for (int s = 0; s < SS; ++s) mx = fmaxf(mx, sc[tid][s]);
    float sm = 0.f;
    for (int s = 0; s < SS; ++s) sm += __expf(sc[tid][s] - mx);
    shmax[tid] = mx; shsum[tid] = sm;
  }
  __syncthreads();
  float asum = 0.f;
  for (int h = 0; h < HH; ++h) {
    float a = __expf(sc[h][tid] - shmax[h]) / shsum[h];
    sc[h][tid] = a;
    asum += a;
  }
  attnw[(size_t)bp * SS + tid] = asum * (1.0f / HH);
  __syncthreads();
  #pragma unroll
  for (int i = 0; i < DD / 128; ++i) {
    int d = tid + i * 128;
    int h = d >> 6;
    float acc = 0.f;
    for (int s = 0; s < SS; ++s)
      acc += sc[h][s] * Vt[((size_t)bp * SS + s) * DD + d];
    ctx[(size_t)bp * DD + d] = acc;
  }
}

// ---------------- LayerNorm(out = g*(a+b - mu)/sqrt(var+eps) + be) ---------
__global__ __launch_bounds__(256) void layernorm_kernel(
    const float* __restrict__ A, const float* __restrict__ Bres,
    const float* __restrict__ g, const float* __restrict__ be,
    float* __restrict__ out)
{
  const int row = blockIdx.x, tid = threadIdx.x;
  const float* pa = A + (size_t)row * DD;
  const float* pb = Bres + (size_t)row * DD;
  float x0 = pa[tid] + pb[tid];
  float x1 = pa[tid + 256] + pb[tid + 256];
  __shared__ float s1[256], s2[256];
  s1[tid] = x0 + x1;
  s2[tid] = x0 * x0 + x1 * x1;
  __syncthreads();
  for (int off = 128; off > 0; off >>= 1) {
    if (tid < off) { s1[tid] += s1[tid + off]; s2[tid] += s2[tid + off]; }
    __syncthreads();
  }
  float mean = s1[0] * (1.0f / DD);
  float var  = s2[0] * (1.0f / DD) - mean * mean;
  float rstd = rsqrtf(var + 1e-6f);
  out[(size_t)row * DD + tid]       = g[tid]       * (x0 - mean) * rstd + be[tid];
  out[(size_t)row * DD + tid + 256] = g[tid + 256] * (x1 - mean) * rstd + be[tid + 256];
}

// ---------------- per-row softmax stats over VOC -> particle weight w ------
__global__ __launch_bounds__(256) void row_softmax_w(
    const float* __restrict__ pred, const int* __restrict__ x,
    float* __restrict__ w_out, float* __restrict__ w_ws)
{
  const int row = blockIdx.x;                  // b*P + p
  const float* p = pred + (size_t)row * VOC;
  __shared__ float sm[256];
  float mx = -3.4e38f;
  for (int i = threadIdx.x; i < VOC; i += 256) mx = fmaxf(mx, p[i]);
  sm[threadIdx.x] = mx; __syncthreads();
  for (int off = 128; off > 0; off >>= 1) {
    if (threadIdx.x < off) sm[threadIdx.x] = fmaxf(sm[threadIdx.x], sm[threadIdx.x + off]);
    __syncthreads();
  }
  mx = sm[0]; __syncthreads();
  float sum = 0.f;
  for (int i = threadIdx.x; i < VOC; i += 256) sum += __expf(p[i] - mx);
  sm[threadIdx.x] = sum; __syncthreads();
  for (int off = 128; off > 0; off >>= 1) {
    if (threadIdx.x < off) sm[threadIdx.x] += sm[threadIdx.x + off];
    __syncthreads();
  }
  if (threadIdx.x == 0) {
    const int b = row / PP;
    float wv = __expf(p[x[b]] - mx) / sm[0];
    w_out[row] = wv;
    w_ws[row]  = wv;
  }
}

// ---------------- categorical resampling i_t ~ softmax(w[b,:]) -------------
__global__ __launch_bounds__(192) void sample_kernel(
    const float* __restrict__ w, int* __restrict__ it, unsigned seed)
{
  int idx = threadIdx.x;
  if (idx >= MR) return;
  const int b = idx / PP;
  float wl[PP];
  float mx = -3.4e38f;
  for (int j = 0; j < PP; ++j) { wl[j] = w[b * PP + j]; mx = fmaxf(mx, wl[j]); }
  float sum = 0.f;
  for (int j = 0; j < PP; ++j) { wl[j] = __expf(wl[j] - mx); sum += wl[j]; }
  float u = rng_u01(pcg_hash((unsigned)idx * 2654435761u + seed)) * sum;
  int pick = PP - 1;
  float c = 0.f;
  for (int j = 0; j < PP; ++j) { c += wl[j]; if (u <= c) { pick = j; break; } }
  it[idx] = pick;
}

// ---------------- weighted-average & argmax predictions --------------------
__global__ __launch_bounds__(256) void pred_reduce_kernel(
    const float* __restrict__ pred, const float* __restrict__ w,
    float* __restrict__ avg, float* __restrict__ maxp)
{
  const int b = blockIdx.x;
  const int n = blockIdx.y * 256 + threadIdx.x;
  int am = 0; float best = w[b * PP];
  for (int p = 1; p < PP; ++p) { float v = w[b * PP + p]; if (v > best) { best = v; am = p; } }
  float acc = 0.f;
  for (int p = 0; p < PP; ++p)
    acc += pred[((size_t)(b * PP + p)) * VOC + n] * w[b * PP + p];
  avg[(size_t)b * VOC + n]  = acc;
  maxp[(size_t)b * VOC + n] = pred[((size_t)(b * PP + am)) * VOC + n];
}

// ---------------- I_new = I with slice t <- i_t (emitted as float) ---------
__global__ __launch_bounds__(256) void inew_kernel(
    const int* __restrict__ I, const int* __restrict__ it,
    const int* __restrict__ tptr, float* __restrict__ out)
{
  int idx = blockIdx.x * 256 + threadIdx.x;
  if (idx >= BB * PP * SS) return;
  const int t = *tptr;
  const int s = idx % SS;
  const int bp = idx / SS;
  out[idx] = (float)((s == t) ? it[bp] : I[idx]);
}

// ---------------- resampled z output ---------------------------------------
__global__ __launch_bounds__(256) void zgather_kernel(
    const float* __restrict__ zws, const int* __restrict__ it,
    float* __restrict__ zout)
{
  int idx = blockIdx.x * 256 + threadIdx.x;
  if (idx >= MR * DD) return;
  const int d = idx % DD;
  const int bp = idx / DD;
  const int b = bp / PP;
  zout[idx] = zws[((size_t)(b * PP + it[bp])) * DD + d];
}

// ===========================================================================
extern "C" void kernel_launch(void* const* d_in, const int* in_sizes, int n_in,
                              void* d_out, int out_size, void* d_ws, size_t ws_size,
                              hipStream_t stream)
{
  (void)in_sizes; (void)n_in; (void)out_size; (void)ws_size;

  const float* r    = (const float*)d_in[0];
  const int*   x    = (const int*)  d_in[1];
  const float* Kin  = (const float*)d_in[2];
  const float* Vin  = (const float*)d_in[3];
  const int*   I    = (const int*)  d_in[5];
  const int*   tptr = (const int*)  d_in[6];
  const float* Wq = (const float*)d_in[7];  const float* bq  = (const float*)d_in[8];
  const float* Wk = (const float*)d_in[9];  const float* bk  = (const float*)d_in[10];
  const float* Wv = (const float*)d_in[11]; const float* bv  = (const float*)d_in[12];
  const float* Wo = (const float*)d_in[13]; const float* bo  = (const float*)d_in[14];
  const float* W1 = (const float*)d_in[15]; const float* b1  = (const float*)d_in[16];
  const float* W2 = (const float*)d_in[17]; const float* b2  = (const float*)d_in[18];
  const float* g1 = (const float*)d_in[19]; const float* be1 = (const float*)d_in[20];
  const float* g3 = (const float*)d_in[21]; const float* be3 = (const float*)d_in[22];
  const float* Wvoc = (const float*)d_in[23]; const float* bvoc = (const float*)d_in[24];

  float* out   = (float*)d_out;
  float* out3  = out + OUT3_OFF;
  float* zout  = out + ZOUT_OFF;
  float* avgp  = out + AVG_OFF;
  float* maxp  = out + MAXP_OFF;
  float* epsO  = out + EPS_OFF;
  float* attnw = out + ATTN_OFF;
  float* Kout  = out + KOUT_OFF;
  float* Vout  = out + VOUT_OFF;
  float* wOut  = out + WOUT_OFF;
  float* InewO = out + INEW_OFF;

  float* ws   = (float*)d_ws;
  float* wsQ  = ws + WS_Q;   float* wsK  = ws + WS_K;   float* wsV  = ws + WS_V;
  float* wsC  = ws + WS_CTX; float* wsZ  = ws + WS_Z;   float* wsO1 = ws + WS_O1;
  float* wsF1 = ws + WS_F1;  float* wsP  = ws + WS_PRED;
  float* wsW  = ws + WS_W;
  int*   wsIt = (int*)(ws + WS_IT);
  float* wsF2 = wsQ;  // q dead after attention; reuse for ffn2

  const dim3 g512(DD / 64, MR / 32);       // GEMMs with N=512
  const dim3 gFF1(DFFC / 64, MR / 32);     // N=2048
  const dim3 gVOC(VOC / 64, MR / 32);      // N=32000

  // 1) QKV projections (WMMA f32)
  gemm_f32_wmma<<<g512, 128, 0, stream>>>(r, Wq, bq, wsQ, MR, DD, DD, 0);
  gemm_f32_wmma<<<g512, 128, 0, stream>>>(r, Wk, bk, wsK, MR, DD, DD, 0);
  gemm_f32_wmma<<<g512, 128, 0, stream>>>(r, Wv, bv, wsV, MR, DD, DD, 0);
  // 2) noise on q,k,v (seeds 1..3)
  noise_kernel<<<(MR * DD + 255) / 256, 256, 0, stream>>>(wsQ, nullptr, MR * DD, 1u);
  noise_kernel<<<(MR * DD + 255) / 256, 256, 0, stream>>>(wsK, nullptr, MR * DD, 2u);
  noise_kernel<<<(MR * DD + 255) / 256, 256, 0, stream>>>(wsV, nullptr, MR * DD, 3u);
  // 3) K/V gather along P + insert new k,v at slice t (writes output K,V)
  gather_kv_kernel<<<BB * PP * SS, 128, 0, stream>>>(Kin, Vin, I, tptr, wsK, wsV, Kout, Vout);
  // 4) attention -> ctx, attn_weights
  attention_kernel<<<MR, 128, 0, stream>>>(wsQ, Kout, Vout, wsC, attnw);
  // 5) z = ctx@Wo + bo, then epsilon noise (seed 4, exported)
  gemm_f32_wmma<<<g512, 128, 0, stream>>>(wsC, Wo, bo, wsZ, MR, DD, DD, 0);
  noise_kernel<<<(MR * DD + 255) / 256, 256, 0, stream>>>(wsZ, epsO, MR * DD, 4u);
  // 6) out1 = LN(z + r)
  layernorm_kernel<<<MR, 256, 0, stream>>>(wsZ, r, g1, be1, wsO1);
  // 7) FFN
  gemm_f32_wmma<<<gFF1, 128, 0, stream>>>(wsO1, W1, b1, wsF1, MR, DFFC, DD, 1);
  gemm_f32_wmma<<<g512, 128, 0, stream>>>(wsF1, W2, b2, wsF2, MR, DD, DFFC, 0);
  // 8) out3 = LN(ffn + out1) -> output
  layernorm_kernel<<<MR, 256, 0, stream>>>(wsF2, wsO1, g3, be3, out3);
  // 9) predictions = out3 @ Wout + bout  (dominant GEMM, WMMA)
  gemm_f32_wmma<<<gVOC, 128, 0, stream>>>(out3, Wvoc, bvoc, wsP, MR, VOC, DD, 0);
  // 10) particle weights w = softmax(pred)[x[b]]
  row_softmax_w<<<MR, 256, 0, stream>>>(wsP, x, wOut, wsW);
  // 11) categorical resampling (seed 5)
  sample_kernel<<<1, 192, 0, stream>>>(wsW, wsIt, 5u);
  // 12) weighted-average & argmax predictions
  pred_reduce_kernel<<<dim3(BB, VOC / 256), 256, 0, stream>>>(wsP, wsW, avgp, maxp);
  // 13) I_new and resampled z
  inew_kernel<<<(BB * PP * SS + 255) / 256, 256, 0, stream>>>(I, wsIt, tptr, InewO);
  zgather_kernel<<<(MR * DD + 255) / 256, 256, 0, stream>>>(wsZ, wsIt, zout);
}